// CatmullRomSplines_90082644066929
// MI455X (gfx1250) — compile-verified
//
#include <hip/hip_runtime.h>

#define NBLOCKS 2048
#define TPB     256
#define G       64

typedef __attribute__((ext_vector_type(2))) float v2f;
typedef __attribute__((ext_vector_type(8))) float v8f;

// ---------------------------------------------------------------------------
// Kernel 1: stream points, gather 4x4 control-point tiles from an LDS-resident
// copy of CP_locs (staged with CDNA5 async global->LDS b128 DMA), accumulate
// per-thread SSE, reduce to one partial per block.
// ---------------------------------------------------------------------------
__global__ __launch_bounds__(TPB) void spline_sse_kernel(
    const float* __restrict__ ch1,      // (N,2)
    const float* __restrict__ ch2,      // (N,2)
    const float* __restrict__ cp,       // (64,64,2)
    const int*   __restrict__ idx,      // (N,2)
    float*       __restrict__ partial,  // (gridDim.x)
    int n)
{
    __shared__ __attribute__((aligned(16))) float lds_cp[G * G * 2]; // 32 KB
    __shared__ float wsum[TPB / 32];

    const int t = threadIdx.x;

    // ---- Stage CP_locs (32768 B) into LDS via async copies (ASYNCcnt path).
    // 8 iterations x 256 threads x 16 B = 32 KB.
    {
        unsigned ldsBase = (unsigned)(unsigned long long)(&lds_cp[0]);
#pragma unroll
        for (int it = 0; it < (G * G * 2 * 4) / (TPB * 16); ++it) {
            unsigned byteOff = (unsigned)((it * TPB + t) * 16);
            unsigned ldsAddr = ldsBase + byteOff;
            asm volatile("global_load_async_to_lds_b128 %0, %1, %2"
                         :: "v"(ldsAddr), "v"(byteOff), "s"(cp)
                         : "memory");
        }
        asm volatile("s_wait_asynccnt 0x0" ::: "memory");
    }
    __syncthreads();

    const float2* __restrict__ ch1v = (const float2*)ch1;
    const float2* __restrict__ ch2v = (const float2*)ch2;
    const int2*   __restrict__ idxv = (const int2*)idx;

    float acc = 0.0f;
    const int stride = gridDim.x * TPB;
    for (int i = blockIdx.x * TPB + t; i < n; i += stride) {
        float2 c1 = ch1v[i];
        float2 c2 = ch2v[i];
        int2   id = idxv[i];

        float x = c2.x - floorf(c2.x);
        float y = c2.y - floorf(c2.y);

        // Catmull-Rom basis: w = (x^3,x^2,x,1) @ A  (Horner / FMA form)
        float wx0 = x * fmaf(x, fmaf(x, -0.5f, 1.0f), -0.5f);
        float wx1 = fmaf(x * x, fmaf(x, 1.5f, -2.5f), 1.0f);
        float wx2 = x * fmaf(x, fmaf(x, -1.5f, 2.0f), 0.5f);
        float wx3 = x * x * fmaf(x, 0.5f, -0.5f);

        float wy0 = y * fmaf(y, fmaf(y, -0.5f, 1.0f), -0.5f);
        float wy1 = fmaf(y * y, fmaf(y, 1.5f, -2.5f), 1.0f);
        float wy2 = y * fmaf(y, fmaf(y, -1.5f, 2.0f), 0.5f);
        float wy3 = y * y * fmaf(y, 0.5f, -0.5f);

        const float wxi[4] = {wx0, wx1, wx2, wx3};

        // Gather 4x4x2 tile starting at (id.x-1, id.y-1) from LDS.
        int r0 = (id.x - 1) * G + (id.y - 1);
        float s0 = 0.0f, s1 = 0.0f;
#pragma unroll
        for (int ii = 0; ii < 4; ++ii) {
            const float2* row = (const float2*)&lds_cp[(r0 + ii * G) * 2];
            float2 q0 = row[0];
            float2 q1 = row[1];
            float2 q2 = row[2];
            float2 q3 = row[3];
            float rx = fmaf(wy0, q0.x, fmaf(wy1, q1.x, fmaf(wy2, q2.x, wy3 * q3.x)));
            float ry = fmaf(wy0, q0.y, fmaf(wy1, q1.y, fmaf(wy2, q2.y, wy3 * q3.y)));
            s0 = fmaf(wxi[ii], rx, s0);
            s1 = fmaf(wxi[ii], ry, s1);
        }

        float d0 = c1.x - s0;
        float d1 = c1.y - s1;
        acc = fmaf(d0, d0, fmaf(d1, d1, acc));
    }

    // ---- Deterministic block reduction (wave32 shuffle tree -> LDS).
#pragma unroll
    for (int off = 16; off > 0; off >>= 1)
        acc += __shfl_down(acc, off);
    if ((t & 31) == 0) wsum[t >> 5] = acc;
    __syncthreads();
    if (t == 0) {
        float b = 0.0f;
#pragma unroll
        for (int w = 0; w < TPB / 32; ++w) b += wsum[w];
        partial[blockIdx.x] = b;
    }
}

// ---------------------------------------------------------------------------
// Kernel 2: one wave sums the per-block partials in fixed order, then does the
// 32-lane cross-lane reduction with V_WMMA_F32_16X16X4_F32:
//   A(16x4): lane L<16 holds {K0,K1}={s_L,0}, lane L+16 holds {K2,K3}={s_L+16,0}
//   B(4x16): all ones  =>  D[m][n] = rowsum(A)[m]; every column of D sums to
//   the grand total. Column 0 lives in lane 0 (rows 0..7) + lane 16 (rows 8..15).
// ---------------------------------------------------------------------------
__global__ __launch_bounds__(32) void final_reduce_kernel(
    const float* __restrict__ partial, float* __restrict__ out, int nb)
{
    int lane = threadIdx.x;
    float s = 0.0f;
    for (int i = lane; i < nb; i += 32) s += partial[i];

#if __has_builtin(__builtin_amdgcn_wmma_f32_16x16x4_f32)
    v2f a; a.x = s;    a.y = 0.0f;
    v2f b; b.x = 1.0f; b.y = 1.0f;
    v8f c = {};
    v8f d = __builtin_amdgcn_wmma_f32_16x16x4_f32(
        /*neg_a=*/false, a, /*neg_b=*/false, b,
        /*c_mod=*/(short)0, c, /*reuse_a=*/false, /*reuse_b=*/false);
    float r = 0.0f;
#pragma unroll
    for (int k = 0; k < 8; ++k) r += d[k];
    float r16 = __shfl(r, 16);
    if (lane == 0) out[0] = r + r16;
#else
#pragma unroll
    for (int off = 16; off > 0; off >>= 1) s += __shfl_down(s, off);
    if (lane == 0) out[0] = s;
#endif
}

extern "C" void kernel_launch(void* const* d_in, const int* in_sizes, int n_in,
                              void* d_out, int out_size, void* d_ws, size_t ws_size,
                              hipStream_t stream) {
    const float* ch1 = (const float*)d_in[0];   // (N,2) f32
    const float* ch2 = (const float*)d_in[1];   // (N,2) f32
    const float* cp  = (const float*)d_in[2];   // (64,64,2) f32
    const int*   idx = (const int*)d_in[3];     // (N,2) i32

    int n = in_sizes[0] / 2;                    // N points
    float* partial = (float*)d_ws;              // NBLOCKS floats of scratch

    spline_sse_kernel<<<NBLOCKS, TPB, 0, stream>>>(ch1, ch2, cp, idx, partial, n);
    final_reduce_kernel<<<1, 32, 0, stream>>>(partial, (float*)d_out, NBLOCKS);
}